// computeMaskedOutput_fixed_class_55490977464601
// MI455X (gfx1250) — compile-verified
//
#include <hip/hip_runtime.h>

// Problem sizes fixed by the reference
constexpr int B    = 128;
constexpr int H    = 14;
constexpr int W    = 14;
constexpr int C    = 512;
constexpr int NCLS = 200;
constexpr int HWp  = H * W;                         // 196
constexpr int TPW  = W + 1;                         // 15
constexpr int TP_ELEMS = (H + 1) * (W + 1) * HWp;   // 44100 floats = 176,400 B
constexpr int TP_VEC16 = TP_ELEMS / 4;              // 11025 x 16-byte chunks
constexpr int CV   = C / 4;                         // 128 float4 per spatial row
constexpr long long NTOT  = (long long)B * HWp * C; // elements per output tensor
constexpr long long NTOT4 = NTOT / 4;               // float4 per output tensor

__global__ __launch_bounds__(128, 1)
void fused_tpl_mask_kernel(const float* __restrict__ x,
                           const float* __restrict__ gt,
                           const float* __restrict__ t_p,
                           const float* __restrict__ ct,
                           float* __restrict__ out)
{
    // 176.4 KB template codebook staged in LDS (gfx1250: up to 320 KB / WGP)
    __shared__ __align__(16) float s_tp[TP_ELEMS];
    __shared__ float s_gv[128];
    __shared__ int   s_gi[128];
    __shared__ int   s_cls;

    const int tid  = threadIdx.x;                   // 0..127, owns channels 4*tid..4*tid+3
    const int b    = blockIdx.x >> 1;               // batch
    const int half = blockIdx.x & 1;                // spatial output half (0..97 / 98..195)

    // ---- 1) async-stage the whole positional template codebook into LDS ----
    // GLOBAL_LOAD_ASYNC_TO_LDS_B128 (GVS mode): saddr = t_p base, vaddr = byte
    // offset, vdst = LDS byte address. Tracked by ASYNCcnt, no VGPR data path.
    {
        const unsigned lds_base = (unsigned)(size_t)(void*)s_tp;
        for (int i = tid; i < TP_VEC16; i += 128) {
            unsigned lds  = lds_base + (unsigned)(i * 16);
            unsigned goff = (unsigned)(i * 16);
            asm volatile("global_load_async_to_lds_b128 %0, %1, %2"
                         :: "v"(lds), "v"(goff), "s"(t_p)
                         : "memory");
        }
    }

    // ---- 2) class id: argmax over gt[b, 0:200] (first-occurrence ties) ----
    {
        const float* gb = gt + (size_t)b * NCLS;
        float v1 = (tid < NCLS) ? gb[tid] : -__builtin_inff();
        int   i1 = tid;
        if (tid + 128 < NCLS) {
            float v2 = gb[tid + 128];
            if (v2 > v1) { v1 = v2; i1 = tid + 128; }   // strict > keeps lower idx
        }
        s_gv[tid] = v1;
        s_gi[tid] = i1;
        __syncthreads();
        for (int s = 64; s > 0; s >>= 1) {
            if (tid < s) {
                float a = s_gv[tid], bb = s_gv[tid + s];
                int   ia = s_gi[tid], ib = s_gi[tid + s];
                if (bb > a || (bb == a && ib < ia)) { s_gv[tid] = bb; s_gi[tid] = ib; }
            }
            __syncthreads();
        }
        if (tid == 0) s_cls = s_gi[0];
    }

    // ---- 3) spatial argmax over 196 positions, float4 (4 channels/lane) ----
    const float4* xb4 = (const float4*)(x + (size_t)b * HWp * C);
    float4 best = xb4[tid];                          // p = 0
    int bp0 = 0, bp1 = 0, bp2 = 0, bp3 = 0;
    for (int p = 1; p < HWp; ++p) {
        float4 v = xb4[(size_t)p * CV + tid];        // 16B/lane, fully coalesced
        if (v.x > best.x) { best.x = v.x; bp0 = p; } // strict > == first occurrence
        if (v.y > best.y) { best.y = v.y; bp1 = p; }
        if (v.z > best.z) { best.z = v.z; bp2 = p; }
        if (v.w > best.w) { best.w = v.w; bp3 = p; }
    }
    // all-zero channel -> sentinel template t_p[H, W]
    const int sent = (H * TPW + W) * HWp;
    const int tb0 = (best.x == 0.0f) ? sent : ((bp0 / W) * TPW + (bp0 % W)) * HWp;
    const int tb1 = (best.y == 0.0f) ? sent : ((bp1 / W) * TPW + (bp1 % W)) * HWp;
    const int tb2 = (best.z == 0.0f) ? sent : ((bp2 / W) * TPW + (bp2 % W)) * HWp;
    const int tb3 = (best.w == 0.0f) ? sent : ((bp3 / W) * TPW + (bp3 % W)) * HWp;

    // ---- 4) wait for the async LDS staging, then make it WG-visible ----
    asm volatile("s_wait_asynccnt 0" ::: "memory");
    __syncthreads();

    const int cls = s_cls;
    const float4* ctb4 = (const float4*)(ct + (size_t)cls * HWp * C);
    __builtin_prefetch(ctb4 + tid, 0, 0);            // global_prefetch_b8

    // ---- 5) fused outputs: masked | x | templates; this block writes its
    //         98-position half with b128 stores (maximizes bytes in flight
    //         per STOREcnt slot: 512B/lane-quad vs 128B scalar) ----
    float4* om = (float4*)out;
    float4* ox = om + NTOT4;
    float4* ot = om + 2 * NTOT4;
    const int pstart = half * (HWp / 2);
    const int pend   = pstart + (HWp / 2);
    for (int p = pstart; p < pend; ++p) {
        if (p + 4 < pend)
            __builtin_prefetch(ctb4 + (size_t)(p + 4) * CV + tid, 0, 0);
        const size_t idx = ((size_t)b * HWp + p) * CV + tid;
        const float4 xv = xb4[(size_t)p * CV + tid]; // L2 hit (x fits in 192MB L2)
        float4 tl;                                   // LDS gather, 4 templates/lane
        tl.x = s_tp[tb0 + p];
        tl.y = s_tp[tb1 + p];
        tl.z = s_tp[tb2 + p];
        tl.w = s_tp[tb3 + p];
        const float4 cv = ctb4[(size_t)p * CV + tid];
        float4 m;
        m.x = fmaxf(xv.x * tl.x, 0.0f) * cv.x;
        m.y = fmaxf(xv.y * tl.y, 0.0f) * cv.y;
        m.z = fmaxf(xv.z * tl.z, 0.0f) * cv.z;
        m.w = fmaxf(xv.w * tl.w, 0.0f) * cv.w;
        om[idx] = m;
        ox[idx] = xv;
        ot[idx] = tl;
    }
}

extern "C" void kernel_launch(void* const* d_in, const int* in_sizes, int n_in,
                              void* d_out, int out_size, void* d_ws, size_t ws_size,
                              hipStream_t stream) {
    (void)in_sizes; (void)n_in; (void)out_size; (void)d_ws; (void)ws_size;
    const float* x   = (const float*)d_in[0];
    const float* gt  = (const float*)d_in[1];
    const float* t_p = (const float*)d_in[2];
    const float* ct  = (const float*)d_in[3];
    float* out = (float*)d_out;

    hipLaunchKernelGGL(fused_tpl_mask_kernel,
                       dim3(2 * B), dim3(128), 0, stream,
                       x, gt, t_p, ct, out);
}